// HierarchicalStateSpaceModel_66597762892286
// MI455X (gfx1250) — compile-verified
//
#include <hip/hip_runtime.h>
#include <hip/hip_bf16.h>

// ---------------------------------------------------------------------------
// HierarchicalStateSpaceModel forward for MI455X (gfx1250, wave32, WMMA).
//
//  * Token-parallel matmuls -> V_WMMA_F32_16X16X4_F32 (exact f32). One wave
//    computes a 16(M)x64(N) strip (4 v8f accums); 256-thr block covers 128x64.
//    Inner loop is branch-free (shift-padded rows read a zero row) and
//    software-pipelined: next k-step's A/B fragments are loaded into distinct
//    registers before the current step's WMMAs issue, so vmem latency is
//    hidden behind matrix ops instead of a full s_wait_loadcnt 0 per WMMA.
//  * GRU: gi = x@Wi+bi hoisted into two big WMMA GEMMs; the sequential h@Wh
//    part is independent per batch row -> 8 workgroups (16 batch rows x 2
//    directions) run the 512-step scan with h in LDS (ds_load_b64 -> v_wmma),
//    same pipelined inner loop.
//  * N=2 heads, VQ argmin, astep recurrence: plain VALU kernels.
//
// Input order assumption: setup_inputs() dict insertion order, params dict
// flattened in insertion order (conv params appended last). seq_size (510) /
// init_size (1) are compile-time constants.
// ---------------------------------------------------------------------------

typedef float v2f __attribute__((ext_vector_type(2)));
typedef float v8f __attribute__((ext_vector_type(8)));

#define WMMA_F32X4(a, b, c) \
  __builtin_amdgcn_wmma_f32_16x16x4_f32(false, (a), false, (b), (short)0, (c), false, false)

namespace {
constexpr int B_BATCH = 64;
constexpr int T_SEQ   = 512;
constexpr int H       = 256;
constexpr int H3      = 768;
constexpr int LN      = 16;
constexpr int SEQ     = 510;   // seq_size
constexpr int INIT    = 1;     // init_size
constexpr size_t NT   = (size_t)B_BATCH * T_SEQ;   // 32768 token rows (b*T+t)
constexpr size_t NS   = (size_t)SEQ * B_BATCH;     // 32640 slice rows (t*B+b)
}

__device__ __forceinline__ v2f ld_a2(const float* p) {
  const float2 t = *(const float2*)p;               // b64 load (global or LDS)
  v2f a; a.x = t.x; a.y = t.y; return a;
}
__device__ __forceinline__ void ld_b4(const float* wp, int ldw, v2f b[4]) {
#pragma unroll
  for (int j = 0; j < 4; ++j) { b[j].x = wp[j * 16]; b[j].y = wp[(size_t)ldw + j * 16]; }
}

// Software-pipelined K-chunk: 4 N-tiles, K rows of A starting at ap, W rows
// starting at wp. Returns wp advanced past the chunk (for concat sources).
__device__ __forceinline__ const float* gemm_chunk4(const float* ap, const float* wp,
                                                    size_t wstep, int ldw, int K,
                                                    v8f acc[4]) {
  v2f a_c = ld_a2(ap);
  v2f b_c[4]; ld_b4(wp, ldw, b_c);
  for (int k = 4; k < K; k += 4) {
    ap += 4; wp += wstep;
    __builtin_prefetch(wp + wstep * 4, 0, 1);       // stream W ahead
    v2f a_n = ld_a2(ap);                            // issue next-step loads
    v2f b_n[4]; ld_b4(wp, ldw, b_n);
#pragma unroll
    for (int j = 0; j < 4; ++j) acc[j] = WMMA_F32X4(a_c, b_c[j], acc[j]);
    a_c = a_n;
#pragma unroll
    for (int j = 0; j < 4; ++j) b_c[j] = b_n[j];
  }
#pragma unroll
  for (int j = 0; j < 4; ++j) acc[j] = WMMA_F32X4(a_c, b_c[j], acc[j]);
  return wp + wstep;
}

// ---------------------------------------------------------------------------
// Generic wave-WMMA GEMM: C = [A0 | A1] @ W + bias  (optional accum / relu).
//   A0 may be row-shifted within a T=512 sequence (causal conv / shifted_enc)
//   or gathered batch-major->time-major (enc_obs_s). A1 is always row m.
//   Shift-invalid rows read a zero row (zrow) -> branch-free inner loop.
// Block: 256 thr = 8 waves; block tile 128(M) x 64(N); wave tile 16 x 64.
// M must be a multiple of 128, N a multiple of 64 (true for all call sites).
// ---------------------------------------------------------------------------
__global__ __launch_bounds__(256)
void wmma_gemm(const float* __restrict__ A0, int K0, int shift0,
               int a0_bmajor, int a0_toff,
               const float* __restrict__ A1, int K1,
               const float* __restrict__ W, int ldw,
               const float* __restrict__ bias,
               const float* __restrict__ zrow,
               float* __restrict__ C, int ldc, int out_bmajor,
               int flags /* bit0: relu, bit1: accumulate into C */)
{
  const int lane = threadIdx.x & 31;
  const int wave = threadIdx.x >> 5;
  const int half = lane >> 4;     // 0: lanes 0-15 (K=0,1), 1: lanes 16-31 (K=2,3)
  const int l    = lane & 15;
  const int m0   = blockIdx.y * 128 + wave * 16;
  const int n0   = blockIdx.x * 64;
  const int m    = m0 + l;        // this lane's logical A row

  // Per-lane A0 row pointer; invalid shifted rows point at the zero row.
  const float* aptr;
  if (a0_bmajor) {
    const int t = m >> 6, b = m & 63;                 // m = t*64 + b (time-major)
    aptr = A0 + (size_t)(b * T_SEQ + t + a0_toff) * K0;
  } else {
    const int t = m & (T_SEQ - 1);                    // m = b*512 + t (batch-major)
    aptr = (t + shift0 >= 0) ? (A0 + (size_t)(m + shift0) * K0) : zrow;
  }
  aptr += 2 * half;

  const float* wp = W + (size_t)(2 * half) * ldw + n0 + l;
  const size_t wstep = (size_t)ldw * 4;

  v8f acc[4];
#pragma unroll
  for (int j = 0; j < 4; ++j)
#pragma unroll
    for (int r = 0; r < 8; ++r) acc[j][r] = 0.0f;

  wp = gemm_chunk4(aptr, wp, wstep, ldw, K0, acc);
  if (A1)                                             // concat second source
    gemm_chunk4(A1 + (size_t)m * K1 + 2 * half, wp, wstep, ldw, K1, acc);

  // Epilogue. C/D layout: lane half h, VGPR r -> row m0 + r + 8h, col n0+j*16+l.
#pragma unroll
  for (int j = 0; j < 4; ++j) {
    const int col = n0 + j * 16 + l;
    const float bv = bias ? bias[col] : 0.0f;
#pragma unroll
    for (int r = 0; r < 8; ++r) {
      const int mr = m0 + r + 8 * half;
      const size_t orow = out_bmajor ? ((size_t)(mr & 63) * SEQ + (mr >> 6))
                                     : (size_t)mr;
      const size_t off = orow * (size_t)ldc + col;
      float v = acc[j][r] + bv;
      if (flags & 2) v += C[off];
      if (flags & 1) v = v > 0.0f ? v : 0.0f;
      C[off] = v;
    }
  }
}

// ---------------------------------------------------------------------------
// Bidirectional GRU scan. gi (input projections) precomputed. Each block owns
// 16 batch rows for one direction; h lives in LDS; per step one 16x768 WMMA
// GEMM (A via ds_load_b64, pipelined) + fused gates. Writes habs time-major
// (T, B, 256). grid = (4, 2): x = batch quarter, y = dir (0 fwd, 1 bwd).
// ---------------------------------------------------------------------------
__global__ __launch_bounds__(256)
void gru_scan(const float* __restrict__ gi_f, const float* __restrict__ gi_b,
              const float* __restrict__ whf,  const float* __restrict__ whb,
              const float* __restrict__ bhf,  const float* __restrict__ bhb,
              float* __restrict__ abs_f,      float* __restrict__ abs_b)
{
  const int rev = blockIdx.y;
  const float* gi = rev ? gi_b : gi_f;
  const float* wh = rev ? whb  : whf;
  const float* bh = rev ? bhb  : bhf;
  float* habs     = rev ? abs_b : abs_f;
  const int b0 = blockIdx.x * 16;

  __shared__ float sh_h[16 * 256];   // hidden state (16 rows)
  __shared__ float sh_g[16 * 768];   // gh = h @ Wh + bh

  for (int i = threadIdx.x; i < 16 * 256; i += 256) sh_h[i] = 0.0f;
  __syncthreads();

  const int lane = threadIdx.x & 31;
  const int wave = threadIdx.x >> 5;
  const int half = lane >> 4;
  const int l    = lane & 15;
  const int c    = threadIdx.x;      // gate-phase column 0..255

  const float* wp_base = wh + (size_t)(2 * half) * 768 + wave * 96 + l;
  const float* hp_base = &sh_h[l * 256 + 2 * half];
  constexpr size_t WSTEP = (size_t)4 * 768;

  for (int s = 0; s < T_SEQ; ++s) {
    const int t = rev ? (T_SEQ - 1 - s) : s;

    // -- phase 1: gh = h @ Wh  (M=16, N=768 -> 48 tiles, 6 per wave),
    //    software-pipelined (LDS A + 12 global B loads ahead of 6 WMMAs) --
    v8f acc[6];
#pragma unroll
    for (int j = 0; j < 6; ++j)
#pragma unroll
      for (int r = 0; r < 8; ++r) acc[j][r] = 0.0f;

    const float* wq = wp_base;
    const float* hp = hp_base;
    v2f a_c = ld_a2(hp);
    v2f b_c[6];
#pragma unroll
    for (int j = 0; j < 6; ++j) { b_c[j].x = wq[j * 16]; b_c[j].y = wq[768 + j * 16]; }
    for (int k = 4; k < 256; k += 4) {
      hp += 4; wq += WSTEP;
      v2f a_n = ld_a2(hp);
      v2f b_n[6];
#pragma unroll
      for (int j = 0; j < 6; ++j) { b_n[j].x = wq[j * 16]; b_n[j].y = wq[768 + j * 16]; }
#pragma unroll
      for (int j = 0; j < 6; ++j) acc[j] = WMMA_F32X4(a_c, b_c[j], acc[j]);
      a_c = a_n;
#pragma unroll
      for (int j = 0; j < 6; ++j) b_c[j] = b_n[j];
    }
#pragma unroll
    for (int j = 0; j < 6; ++j) acc[j] = WMMA_F32X4(a_c, b_c[j], acc[j]);

#pragma unroll
    for (int j = 0; j < 6; ++j) {
      const int col = wave * 96 + j * 16 + l;
      const float bv = bh[col];
#pragma unroll
      for (int r = 0; r < 8; ++r)
        sh_g[(r + 8 * half) * 768 + col] = acc[j][r] + bv;
    }
    __syncthreads();

    // -- phase 2: gates; thread owns column c for all 16 rows --
    float hnew[16];
#pragma unroll
    for (int rr = 0; rr < 16; ++rr) {
      const size_t gr = ((size_t)(b0 + rr) * T_SEQ + t) * H3;
      const float ir  = gi[gr + c];
      const float iz  = gi[gr + 256 + c];
      const float inn = gi[gr + 512 + c];
      const float hr  = sh_g[rr * 768 + c];
      const float hz  = sh_g[rr * 768 + 256 + c];
      const float hn  = sh_g[rr * 768 + 512 + c];
      const float rg = 1.0f / (1.0f + __expf(-(ir + hr)));
      const float zg = 1.0f / (1.0f + __expf(-(iz + hz)));
      const float n  = tanhf(inn + rg * hn);
      hnew[rr] = (1.0f - zg) * n + zg * sh_h[rr * 256 + c];
    }
    __syncthreads();
#pragma unroll
    for (int rr = 0; rr < 16; ++rr) {
      sh_h[rr * 256 + c] = hnew[rr];
      habs[((size_t)t * B_BATCH + (b0 + rr)) * H + c] = hnew[rr];
    }
    __syncthreads();
  }
}

// ---------------------------------------------------------------------------
// Boundary head (N=2) + argmax straight-through + edge masking.
// ---------------------------------------------------------------------------
__global__ void bnd_bdata(const float* __restrict__ x, const float* __restrict__ Wb,
                          const float* __restrict__ bb,
                          float* __restrict__ out_post, float* __restrict__ out_bdata,
                          float* __restrict__ readb, float* __restrict__ copyb)
{
  const int m = blockIdx.x * blockDim.x + threadIdx.x;
  if (m >= (int)NT) return;
  const float* xr = x + (size_t)m * H;
  float s0 = bb[0], s1 = bb[1];
#pragma unroll 4
  for (int k = 0; k < H; ++k) { const float xv = xr[k]; s0 += xv * Wb[2 * k]; s1 += xv * Wb[2 * k + 1]; }
  out_post[2 * m] = s0; out_post[2 * m + 1] = s1;
  // bdata value == one_hot(argmax(softmax(s))) (straight-through); tie -> idx 0.
  const float hard0 = (s0 >= s1) ? 1.0f : 0.0f;
  const float hard1 = 1.0f - hard0;
  const int t = m & (T_SEQ - 1);
  const bool edge = (t <= INIT) || (t >= T_SEQ - INIT);
  const float rd = edge ? 1.0f : hard0;
  const float cp = edge ? 0.0f : hard1;
  out_bdata[2 * m] = rd; out_bdata[2 * m + 1] = cp;
  readb[m] = rd; copyb[m] = cp;
}

// ---------------------------------------------------------------------------
// VQ: nearest code, one-hot (transposed to (B,SEQ,16)), zq, loss accumulation.
// ---------------------------------------------------------------------------
__global__ __launch_bounds__(256)
void vq_kernel(const float* __restrict__ ze, const float* __restrict__ cb,
               float* __restrict__ zvq, float* __restrict__ onehot_out,
               float* __restrict__ accum)
{
  __shared__ float scb[LN * H];
  for (int i = threadIdx.x; i < LN * H; i += 256) scb[i] = cb[i];
  __syncthreads();
  const int m = blockIdx.x * 256 + threadIdx.x;
  if (m >= (int)NS) return;
  const float* zr = ze + (size_t)m * H;
  int best = 0; float bd = 3.4e38f;
  for (int j = 0; j < LN; ++j) {
    float d = 0.0f;
#pragma unroll 4
    for (int k = 0; k < H; ++k) { const float df = zr[k] - scb[j * H + k]; d += df * df; }
    if (d < bd) { bd = d; best = j; }
  }
  const int t = m >> 6, b = m & 63;
  float* oh = onehot_out + ((size_t)b * SEQ + t) * LN;
  for (int j = 0; j < LN; ++j) oh[j] = (j == best) ? 1.0f : 0.0f;
  float* zq = zvq + (size_t)m * H;
  const float* cbr = scb + best * H;
  for (int k = 0; k < H; ++k) zq[k] = cbr[k];
  atomicAdd(accum, bd);   // sum (ze - zq)^2
}

// ---------------------------------------------------------------------------
// astep recurrence: s_t = r_t * z_t + c_t * s_{t-1}; parallel over (b, h).
// ---------------------------------------------------------------------------
__global__ void astep_scan(const float* __restrict__ zvq, const float* __restrict__ readb,
                           const float* __restrict__ copyb, float* __restrict__ abs_state)
{
  const int tid = blockIdx.x * blockDim.x + threadIdx.x;
  if (tid >= B_BATCH * H) return;
  const int b = tid >> 8, c = tid & 255;
  float s = 0.0f;
  for (int t = 0; t < SEQ; ++t) {
    const float r  = readb[b * T_SEQ + t + INIT];
    const float cp = copyb[b * T_SEQ + t + INIT];
    const float z  = zvq[((size_t)t * B_BATCH + b) * H + c];
    s = r * z + cp * s;
    abs_state[((size_t)t * B_BATCH + b) * H + c] = s;
  }
}

// ---------------------------------------------------------------------------
// Prior head (N=2), output transposed to (B, SEQ, 2).
// ---------------------------------------------------------------------------
__global__ void prior_kernel(const float* __restrict__ of, const float* __restrict__ Wpb,
                             const float* __restrict__ bpb, float* __restrict__ out)
{
  const int m = blockIdx.x * blockDim.x + threadIdx.x;
  if (m >= (int)NS) return;
  const float* xr = of + (size_t)m * H;
  float s0 = bpb[0], s1 = bpb[1];
#pragma unroll 4
  for (int k = 0; k < H; ++k) { const float xv = xr[k]; s0 += xv * Wpb[2 * k]; s1 += xv * Wpb[2 * k + 1]; }
  const int t = m >> 6, b = m & 63;
  out[((size_t)b * SEQ + t) * 2]     = s0;
  out[((size_t)b * SEQ + t) * 2 + 1] = s1;
}

__global__ void init_ws(float* __restrict__ zrow, float* __restrict__ accum)
{
  const int i = blockIdx.x * blockDim.x + threadIdx.x;
  if (i < 512) zrow[i] = 0.0f;
  if (i == 0) *accum = 0.0f;
}
__global__ void fin_loss(const float* __restrict__ acc, float* __restrict__ out)
{
  if (threadIdx.x == 0 && blockIdx.x == 0)
    *out = 1.25f * (*acc) / ((float)NS * (float)H);   // mse + 0.25*mse
}

// ---------------------------------------------------------------------------
extern "C" void kernel_launch(void* const* d_in, const int* in_sizes, int n_in,
                              void* d_out, int out_size, void* d_ws, size_t ws_size,
                              hipStream_t stream) {
  // ---- inputs (setup_inputs dict insertion order; params in insertion order)
  const float* obs    = (const float*)d_in[0];   // (64,512,512)
  const float* act    = (const float*)d_in[1];   // (64,512,32)
  const float* W_act  = (const float*)d_in[2];   // (32,256)
  const float* b_act  = (const float*)d_in[3];
  const float* W_enc  = (const float*)d_in[4];   // (512,256)
  const float* b_enc  = (const float*)d_in[5];
  const float* W_comb = (const float*)d_in[6];   // (512,256)
  const float* b_comb = (const float*)d_in[7];
  const float* W_bnd  = (const float*)d_in[8];   // (256,2)
  const float* b_bnd  = (const float*)d_in[9];
  const float* gfi_w  = (const float*)d_in[10];  // (256,768)
  const float* gfh_w  = (const float*)d_in[11];
  const float* gfi_b  = (const float*)d_in[12];
  const float* gfh_b  = (const float*)d_in[13];
  const float* gbi_w  = (const float*)d_in[14];
  const float* gbh_w  = (const float*)d_in[15];
  const float* gbi_b  = (const float*)d_in[16];
  const float* gbh_b  = (const float*)d_in[17];
  const float* W_vq   = (const float*)d_in[18];  // (512,256)
  const float* b_vq   = (const float*)d_in[19];
  const float* codebook = (const float*)d_in[20];// (16,256)
  const float* W_af   = (const float*)d_in[21];  // (512,256)
  const float* b_af   = (const float*)d_in[22];
  const float* W_po   = (const float*)d_in[23];  // (512,256)
  const float* b_po   = (const float*)d_in[24];
  const float* W_of   = (const float*)d_in[25];  // (384,256)
  const float* b_of   = (const float*)d_in[26];
  const float* W_pb   = (const float*)d_in[27];  // (256,2)
  const float* b_pb   = (const float*)d_in[28];
  const float* W_dec  = (const float*)d_in[29];  // (256,512)
  const float* b_dec  = (const float*)d_in[30];
  const float* conv_w0 = (const float*)d_in[31]; // (3,256,256) [tap][in][out]
  const float* conv_b0 = (const float*)d_in[32];
  const float* conv_w1 = (const float*)d_in[33];
  const float* conv_b1 = (const float*)d_in[34];
  // d_in[35]=seq_size(510), d_in[36]=init_size(1): compile-time constants here.

  // ---- workspace layout (floats); later regions reuse freed earlier ones
  float* ws = (float*)d_ws;
  const size_t O_ENC_OBS = 0;                        // NT*256  enc_obs (live to po)
  const size_t O_BUFA    = O_ENC_OBS + NT * 256;     // enc_act -> zvq
  const size_t O_COMB    = O_BUFA   + NT * 256;      // comb -> ze
  const size_t O_XB0     = O_COMB   + NT * 256;      // shifted_comb/conv -> abs_fwd
  const size_t O_XB1     = O_XB0    + NT * 256;      // conv ping -> abs_bwd
  const size_t O_READ    = O_XB1    + NT * 256;      // NT
  const size_t O_COPY    = O_READ   + NT;            // NT
  const size_t O_ACC     = O_COPY   + NT;            // 8 (scalar + pad)
  const size_t O_ZROW    = O_ACC    + 8;             // 512 zero row for shifts
  const size_t O_GIF     = O_ZROW   + 512;           // NT*768  gi fwd -> reuse
  const size_t O_GIB     = O_GIF    + NT * 768;      // NT*768  gi bwd -> reuse
  const size_t O_ABSST   = O_GIF;                    // NS*256
  const size_t O_ABSFEAT = O_GIF + NS * 256;         // NS*256
  const size_t O_OBSST   = O_GIF + 2 * NS * 256;     // NS*128
  const size_t O_OBSFEAT = O_GIB;                    // NS*256
  // total need: O_GIB + NT*768  (~92.3M floats, ~370 MB)
  const float* zrow = ws + O_ZROW;

  // ---- output layout (return-order concat, all f32)
  float* out0 = (float*)d_out;                       // obs_rec        (64,510,512)
  float* out1 = out0 + (size_t)64 * 510 * 512;       // prior_log_alpha(64,510,2)
  float* out2 = out1 + (size_t)64 * 510 * 2;         // post_log_alpha (64,512,2)
  float* out3 = out2 + (size_t)64 * 512 * 2;         // bdata          (64,512,2)
  float* out4 = out3 + (size_t)64 * 512 * 2;         // vq_loss scalar
  float* out5 = out4 + 1;                            // onehot         (64,510,16)

  const dim3 blk(256);
  auto gemm_grid = [](size_t M, int N) { return dim3((unsigned)(N / 64), (unsigned)(M / 128)); };

  // 0) init zero row + loss accumulator
  init_ws<<<dim3(2), blk, 0, stream>>>(ws + O_ZROW, ws + O_ACC);
  // 1) enc_obs = obs @ W_enc + b_enc
  wmma_gemm<<<gemm_grid(NT, 256), blk, 0, stream>>>(
      obs, 512, 0, 0, 0, nullptr, 0, W_enc, 256, b_enc, zrow, ws + O_ENC_OBS, 256, 0, 0);
  // 2) enc_act = act @ W_act + b_act
  wmma_gemm<<<gemm_grid(NT, 256), blk, 0, stream>>>(
      act, 32, 0, 0, 0, nullptr, 0, W_act, 256, b_act, zrow, ws + O_BUFA, 256, 0, 0);
  // 3) comb = [enc_act, enc_obs] @ W_comb + b_comb
  wmma_gemm<<<gemm_grid(NT, 256), blk, 0, stream>>>(
      ws + O_BUFA, 256, 0, 0, 0, ws + O_ENC_OBS, 256, W_comb, 256, b_comb, zrow,
      ws + O_COMB, 256, 0, 0);
  // 4) shifted_comb = [shift(enc_act,-1), enc_obs] @ W_comb + b_comb
  wmma_gemm<<<gemm_grid(NT, 256), blk, 0, stream>>>(
      ws + O_BUFA, 256, -1, 0, 0, ws + O_ENC_OBS, 256, W_comb, 256, b_comb, zrow,
      ws + O_XB0, 256, 0, 0);
  // 5) causal convs (tap dk uses time shift dk-2; accumulate, relu at end)
  for (int layer = 0; layer < 2; ++layer) {
    const float* cw  = layer ? conv_w1 : conv_w0;
    const float* cbv = layer ? conv_b1 : conv_b0;
    float* xin  = ws + (layer ? O_XB1 : O_XB0);
    float* xout = ws + (layer ? O_XB0 : O_XB1);
    for (int dk = 0; dk < 3; ++dk) {
      const int flags = (dk == 0 ? 0 : 2) | (dk == 2 ? 1 : 0);
      wmma_gemm<<<gemm_grid(NT, 256), blk, 0, stream>>>(
          xin, 256, dk - 2, 0, 0, nullptr, 0, cw + (size_t)dk * 256 * 256, 256,
          dk == 0 ? cbv : nullptr, zrow, xout, 256, 0, flags);
    }
  }
  // 6) boundary head + bdata/read/copy (final conv output is in XB0)
  bnd_bdata<<<dim3(NT / 256), blk, 0, stream>>>(
      ws + O_XB0, W_bnd, b_bnd, out2, out3, ws + O_READ, ws + O_COPY);
  // 7) GRU input projections (recurrence-free -> big GEMMs)
  wmma_gemm<<<gemm_grid(NT, 768), blk, 0, stream>>>(
      ws + O_COMB, 256, 0, 0, 0, nullptr, 0, gfi_w, 768, gfi_b, zrow, ws + O_GIF, 768, 0, 0);
  wmma_gemm<<<gemm_grid(NT, 768), blk, 0, stream>>>(
      ws + O_COMB, 256, 0, 0, 0, nullptr, 0, gbi_w, 768, gbi_b, zrow, ws + O_GIB, 768, 0, 0);
  // 8) bidirectional GRU scans (h in LDS; abs_fwd -> XB0, abs_bwd -> XB1, time-major)
  gru_scan<<<dim3(4, 2), blk, 0, stream>>>(
      ws + O_GIF, ws + O_GIB, gfh_w, gbh_w, gfh_b, gbh_b, ws + O_XB0, ws + O_XB1);
  // 9) ze = [abs_fwd[0:510], abs_bwd[1:511]] @ W_vq + b_vq   (time-major rows)
  wmma_gemm<<<gemm_grid(NS, 256), blk, 0, stream>>>(
      ws + O_XB0, 256, 0, 0, 0, ws + O_XB1 + (size_t)B_BATCH * H, 256,
      W_vq, 256, b_vq, zrow, ws + O_COMB, 256, 0, 0);
  // 10) VQ: nearest code, onehot(out5), zq -> zvq(BUFA), loss accumulation
  vq_kernel<<<dim3((unsigned)((NS + 255) / 256)), blk, 0, stream>>>(
      ws + O_COMB, codebook, ws + O_BUFA, out5, ws + O_ACC);
  // 11) astep recurrence -> abs_state
  astep_scan<<<dim3(64), blk, 0, stream>>>(
      ws + O_BUFA, ws + O_READ, ws + O_COPY, ws + O_ABSST);
  // 12) abs_feat = abs_state @ W_af[H:] + b_af
  wmma_gemm<<<gemm_grid(NS, 256), blk, 0, stream>>>(
      ws + O_ABSST, 256, 0, 0, 0, nullptr, 0, W_af + 256 * 256, 256, b_af, zrow,
      ws + O_ABSFEAT, 256, 0, 0);
  // 13) obs_state = [enc_obs_s, abs_feat] @ W_po[:, :128] + b_po[:128]
  //     enc_obs is batch-major -> a0_bmajor gather with toff=INIT
  wmma_gemm<<<gemm_grid(NS, 128), blk, 0, stream>>>(
      ws + O_ENC_OBS, 256, 0, 1, INIT, ws + O_ABSFEAT, 256, W_po, 256, b_po, zrow,
      ws + O_OBSST, 128, 0, 0);
  // 14) obs_feat = obs_state @ W_of[H:] + b_of   (K=128)
  wmma_gemm<<<gemm_grid(NS, 256), blk, 0, stream>>>(
      ws + O_OBSST, 128, 0, 0, 0, nullptr, 0, W_of + 256 * 256, 256, b_of, zrow,
      ws + O_OBSFEAT, 256, 0, 0);
  // 15) prior_log_alpha (N=2) -> out1 (transposed to (B,SEQ,2))
  prior_kernel<<<dim3((unsigned)((NS + 255) / 256)), blk, 0, stream>>>(
      ws + O_OBSFEAT, W_pb, b_pb, out1);
  // 16) obs_rec = obs_feat @ W_dec + b_dec -> out0 transposed to (B,SEQ,512)
  wmma_gemm<<<gemm_grid(NS, 512), blk, 0, stream>>>(
      ws + O_OBSFEAT, 256, 0, 0, 0, nullptr, 0, W_dec, 512, b_dec, zrow, out0, 512, 1, 0);
  // 17) vq_loss = 1.25 * mean((ze - zq)^2) -> out4
  fin_loss<<<dim3(1), dim3(32), 0, stream>>>(ws + O_ACC, out4);

  (void)in_sizes; (void)n_in; (void)out_size; (void)ws_size;
}